// FeatureGraph_41240275976717
// MI455X (gfx1250) — compile-verified
//
#include <hip/hip_runtime.h>
#include <hip/hip_bf16.h>
#include <stdint.h>

#define BGRAPH   512
#define NNODES   64
#define INCH     256
#define EMBED    64
#define TOPK     20
#define EOUT     (BGRAPH * NNODES * TOPK)   // 655360

typedef float v2f __attribute__((ext_vector_type(2)));
typedef float v8f __attribute__((ext_vector_type(8)));

__device__ __forceinline__ float wred_max_f(float v) {
    #pragma unroll
    for (int o = 16; o > 0; o >>= 1) v = fmaxf(v, __shfl_xor(v, o, 32));
    return v;
}
__device__ __forceinline__ float wred_sum_f(float v) {
    #pragma unroll
    for (int o = 16; o > 0; o >>= 1) v += __shfl_xor(v, o, 32);
    return v;
}
__device__ __forceinline__ unsigned long long wred_max_u64(unsigned long long v) {
    #pragma unroll
    for (int o = 16; o > 0; o >>= 1) {
        unsigned long long t = __shfl_xor(v, o, 32);
        v = (t > v) ? t : v;
    }
    return v;
}

// One workgroup (256 threads = 8 waves) per graph.
__global__ __launch_bounds__(256)
void feature_graph_fused(const float* __restrict__ x,
                         const float* __restrict__ Wl, const float* __restrict__ bl,
                         const float* __restrict__ Wr, const float* __restrict__ br,
                         const float* __restrict__ att,
                         int* __restrict__ out_i, int* __restrict__ out_j,
                         float* __restrict__ out_a) {
    __shared__ float s_xl[NNODES][EMBED + 1];
    __shared__ float s_xr[NNODES][EMBED + 1];
    __shared__ float s_al[NNODES][EMBED + 1];
    __shared__ float s_att[EMBED];

    const int g    = blockIdx.x;
    const int tid  = threadIdx.x;
    const int lane = tid & 31;
    const int wv   = tid >> 5;

    if (tid < EMBED) s_att[tid] = att[tid];

    // ---------------- Phase 1: x_l / x_r GEMM via V_WMMA_F32_16X16X4_F32 ---
    // waves 0-3 -> x_l strip rows [wv*16, wv*16+16); waves 4-7 -> x_r.
    {
        const bool right = (wv >= 4);
        const float* W    = right ? Wr : Wl;
        const float* bias = right ? br : bl;
        float (*dst)[EMBED + 1] = right ? s_xr : s_xl;
        const int mi    = wv & 3;
        const int mlane = lane & 15;           // M (A) / N (B,C,D) within tile
        const int khalf = (lane >> 4) << 1;    // K pair selector: 0 or 2

        const float* xrow = x + ((size_t)(g * NNODES + mi * 16 + mlane)) * INCH;

        v8f acc[4];
        #pragma unroll
        for (int ni = 0; ni < 4; ++ni) acc[ni] = (v8f){0,0,0,0,0,0,0,0};

        for (int k0 = 0; k0 < INCH; k0 += 4) {
            const v2f a = *(const v2f*)(xrow + k0 + khalf);        // A: 16x4 fp32
            #pragma unroll
            for (int ni = 0; ni < 4; ++ni) {
                const v2f b = *(const v2f*)(W + (size_t)(ni * 16 + mlane) * INCH
                                              + k0 + khalf);        // B: 4x16 fp32
                acc[ni] = __builtin_amdgcn_wmma_f32_16x16x4_f32(
                    false, a, false, b, (short)0, acc[ni], false, false);
            }
        }
        #pragma unroll
        for (int ni = 0; ni < 4; ++ni) {
            const float bv = bias[ni * 16 + mlane];
            #pragma unroll
            for (int v = 0; v < 8; ++v) {
                const int row = mi * 16 + v + ((lane >> 4) << 3); // C/D layout
                dst[row][ni * 16 + mlane] = acc[ni][v] + bv;
            }
        }
    }
    __syncthreads();

    // ---------------- Phase 2: alpha[i][j] = sum_d att_d*leaky(xl[i,d]+xr[j,d])
    #pragma unroll 1
    for (int it = 0; it < 16; ++it) {
        const int e = it * 256 + tid;
        const int i = e >> 6, j = e & 63;
        float sum = 0.f;
        #pragma unroll 8
        for (int d = 0; d < EMBED; ++d) {
            const float z  = s_xl[i][d] + s_xr[j][d];
            const float lr = fmaxf(z, 0.f) + 0.2f * fminf(z, 0.f);
            sum = fmaf(s_att[d], lr, sum);
        }
        s_al[i][j] = sum;
    }
    __syncthreads();

    // ---------------- Phase 3: softmax (self-loop in denom) + diag-zero + top-20
    #pragma unroll 1
    for (int r = 0; r < 8; ++r) {
        const int i = wv * 8 + r;
        const float a0 = s_al[i][lane];
        const float a1 = s_al[i][lane + 32];
        const float m  = wred_max_f(fmaxf(a0, a1));
        const float e0 = __expf(a0 - m);
        const float e1 = __expf(a1 - m);
        const float inv = 1.0f / wred_sum_f(e0 + e1);
        const float p0 = e0 * inv;
        const float p1 = e1 * inv;

        // key = (float_bits << 6) | (63 - j): positive floats order as uints,
        // low bits implement lower-index-first tie break. Diagonal -> key 0.
        unsigned long long k0 = ((unsigned long long)__float_as_uint(p0) << 6)
                                | (unsigned)(63 - lane);
        unsigned long long k1 = ((unsigned long long)__float_as_uint(p1) << 6)
                                | (unsigned)(63 - (lane + 32));
        if (lane == i)      k0 = 0ull;
        if (lane + 32 == i) k1 = 0ull;

        unsigned long long mykey = 0ull;
        #pragma unroll 1
        for (int t = 0; t < TOPK; ++t) {
            unsigned long long cand = (k0 > k1) ? k0 : k1;
            cand = wred_max_u64(cand);                 // all lanes get winner
            const int jw = 63 - (int)(cand & 63ull);
            if (jw == lane)           k0 = 0ull;       // remove winner
            else if (jw == lane + 32) k1 = 0ull;
            if (lane == t) mykey = cand;               // lane t keeps rank-t entry
        }

        if (lane < TOPK) {
            const int node = g * NNODES + i;
            const int eo   = node * TOPK + lane;
            out_i[eo] = node;
            out_j[eo] = g * NNODES + (63 - (int)(mykey & 63ull));
            out_a[eo] = __uint_as_float((unsigned)(mykey >> 6));
        }
    }
}

extern "C" void kernel_launch(void* const* d_in, const int* in_sizes, int n_in,
                              void* d_out, int out_size, void* d_ws, size_t ws_size,
                              hipStream_t stream) {
    // setup_inputs order: x, edge_index, batch, Wl, bl, Wr, br, att
    const float* x   = (const float*)d_in[0];
    const float* Wl  = (const float*)d_in[3];
    const float* bl  = (const float*)d_in[4];
    const float* Wr  = (const float*)d_in[5];
    const float* br  = (const float*)d_in[6];
    const float* att = (const float*)d_in[7];

    // d_out: new_edge_index [2, EOUT] (int32) then attention [EOUT] (f32)
    int*   out_i = (int*)d_out;
    int*   out_j = out_i + EOUT;
    float* out_a = (float*)d_out + 2 * EOUT;

    (void)in_sizes; (void)n_in; (void)out_size; (void)d_ws; (void)ws_size;
    feature_graph_fused<<<dim3(BGRAPH), dim3(256), 0, stream>>>(
        x, Wl, bl, Wr, br, att, out_i, out_j, out_a);
}